// ComputePairwiseDistances_68650757259978
// MI455X (gfx1250) — compile-verified
//
#include <hip/hip_runtime.h>

typedef __attribute__((ext_vector_type(2))) float v2f;
typedef __attribute__((ext_vector_type(4))) float v4f;
typedef __attribute__((ext_vector_type(8))) float v8f;
typedef __attribute__((ext_vector_type(4))) int   v4i;

#define K_DIM 512
#define BK 16
#define NSLAB (K_DIM / BK)          // 32 K-slabs
#define LDS_STRIDE (BK + 4)         // 20 floats: pad to spread LDS banks
#define SLAB_FLOATS (128 * LDS_STRIDE)

// ---------------------------------------------------------------------------
// CDNA5 async global->LDS copy (ASYNCcnt-tracked DMA, no VGPR round trip).
// Builtin signature (from hipcc diagnostic): param0 = v4i in AS(1) (global
// src, non-const), param1 = v4i in AS(3) (LDS dst), then imm offset, imm cpol.
// ---------------------------------------------------------------------------
__device__ __forceinline__ void async_g2l_b128(void* lds_dst, const void* gsrc) {
#if __has_builtin(__builtin_amdgcn_global_load_async_to_lds_b128)
  __builtin_amdgcn_global_load_async_to_lds_b128(
      (__attribute__((address_space(1))) v4i*)(v4i*)(void*)(size_t)gsrc,
      (__attribute__((address_space(3))) v4i*)(v4i*)lds_dst,
      /*offset=*/0, /*cpol=*/0);
#else
  unsigned lofs = (unsigned)(size_t)(__attribute__((address_space(3))) char*)lds_dst;
  asm volatile("global_load_async_to_lds_b128 %0, %1, off"
               :: "v"(lofs), "v"(gsrc) : "memory");
#endif
}

__device__ __forceinline__ void wait_async0() {
#if __has_builtin(__builtin_amdgcn_s_wait_asynccnt)
  __builtin_amdgcn_s_wait_asynccnt(0);
#else
  asm volatile("s_wait_asynccnt 0x0" ::: "memory");
#endif
}

// ---------------------------------------------------------------------------
// Kernel 1: per-row squared norms for X (queries) and Y (prototypes).
// One wave32 per row; float4 loads; cross-lane reduce with shfl_xor.
// ---------------------------------------------------------------------------
__global__ __launch_bounds__(256)
void row_sqnorm_kernel(const float* __restrict__ X, const float* __restrict__ Y,
                       float* __restrict__ xsq, float* __restrict__ ysq,
                       int nX, int nY) {
  int gwave = (int)((blockIdx.x * blockDim.x + threadIdx.x) >> 5);
  int lane  = (int)(threadIdx.x & 31);
  int total = nX + nY;
  if (gwave >= total) return;

  const float* src = (gwave < nX) ? (X + (size_t)gwave * K_DIM)
                                  : (Y + (size_t)(gwave - nX) * K_DIM);
  float s = 0.0f;
#pragma unroll
  for (int j = 0; j < K_DIM / 128; ++j) {          // 4 x float4 per lane
    v4f v = *(const v4f*)(src + (size_t)(lane + 32 * j) * 4);
    s += v.x * v.x + v.y * v.y + v.z * v.z + v.w * v.w;
  }
#pragma unroll
  for (int off = 16; off > 0; off >>= 1)
    s += __shfl_xor(s, off, 32);

  if (lane == 0) {
    if (gwave < nX) xsq[gwave] = s;
    else            ysq[gwave - nX] = s;
  }
}

// ---------------------------------------------------------------------------
// Kernel 2: out[n][m] = (xsq[n] + ysq[m] - 2 * dot(X[n], Y[m])) / K_DIM
// 128x128 block tile, 8 waves, each wave computes 64x32 as 4x2 wmma tiles.
// K staged through double-buffered LDS via async DMA; slab s+1 DMA overlaps
// slab s WMMA compute.
// ---------------------------------------------------------------------------
__global__ __launch_bounds__(256)
void dist_wmma_kernel(const float* __restrict__ X, const float* __restrict__ Y,
                      const float* __restrict__ xsq, const float* __restrict__ ysq,
                      float* __restrict__ out, int Mcols) {
  __shared__ float Xs[2][SLAB_FLOATS];
  __shared__ float Ys[2][SLAB_FLOATS];

  const int tid  = (int)threadIdx.x;
  const int lane = tid & 31;
  const int wave = tid >> 5;
  const int wr = wave & 1;      // row group: 2 x 64 rows
  const int wc = wave >> 1;     // col group: 4 x 32 cols
  const int lhalf = lane >> 4;  // 0|1 -> K sub-pair / M+8 half
  const int lmod  = lane & 15;  // M (A) / N (B,C,D) index

  const int rowBase = (int)blockIdx.y * 128;  // query rows
  const int colBase = (int)blockIdx.x * 128;  // prototype cols

  v8f acc[4][2];
#pragma unroll
  for (int ti = 0; ti < 4; ++ti)
#pragma unroll
    for (int tj = 0; tj < 2; ++tj)
      acc[ti][tj] = (v8f){0.f, 0.f, 0.f, 0.f, 0.f, 0.f, 0.f, 0.f};

  // Per-thread staging coordinates: 128 rows x 4 float4 per row per matrix,
  // 256 threads -> 2 (row, c4) pairs each.
  int srow[2], sc4[2];
#pragma unroll
  for (int i = 0; i < 2; ++i) {
    int idx = tid + 256 * i;       // 0..511
    srow[i] = idx >> 2;            // 0..127
    sc4[i]  = (idx & 3) << 2;      // 0,4,8,12
  }

  // Prologue: async-stage slab 0 into buffer 0.
#pragma unroll
  for (int i = 0; i < 2; ++i) {
    async_g2l_b128(&Xs[0][srow[i] * LDS_STRIDE + sc4[i]],
                   X + (size_t)(rowBase + srow[i]) * K_DIM + sc4[i]);
    async_g2l_b128(&Ys[0][srow[i] * LDS_STRIDE + sc4[i]],
                   Y + (size_t)(colBase + srow[i]) * K_DIM + sc4[i]);
  }
  wait_async0();
  __syncthreads();

  for (int s = 0; s < NSLAB; ++s) {
    const int cur = s & 1;
    const int nxt = cur ^ 1;

    // Kick off DMA for the next slab into the other buffer (overlaps WMMA).
    if (s + 1 < NSLAB) {
      const int k0n = (s + 1) * BK;
#pragma unroll
      for (int i = 0; i < 2; ++i) {
        async_g2l_b128(&Xs[nxt][srow[i] * LDS_STRIDE + sc4[i]],
                       X + (size_t)(rowBase + srow[i]) * K_DIM + k0n + sc4[i]);
        async_g2l_b128(&Ys[nxt][srow[i] * LDS_STRIDE + sc4[i]],
                       Y + (size_t)(colBase + srow[i]) * K_DIM + k0n + sc4[i]);
      }
    }

    // Compute current slab: 4 k-steps x (4x2) tiles = 32 wmma per wave.
#pragma unroll
    for (int kk = 0; kk < BK; kk += 4) {
      const int kfrag = kk + lhalf * 2;   // even -> 8B-aligned v2f loads
      v2f a[4], b[2];
#pragma unroll
      for (int ti = 0; ti < 4; ++ti)
        a[ti] = *(const v2f*)(&Xs[cur][(wr * 64 + ti * 16 + lmod) * LDS_STRIDE + kfrag]);
#pragma unroll
      for (int tj = 0; tj < 2; ++tj)
        b[tj] = *(const v2f*)(&Ys[cur][(wc * 32 + tj * 16 + lmod) * LDS_STRIDE + kfrag]);
#pragma unroll
      for (int ti = 0; ti < 4; ++ti)
#pragma unroll
        for (int tj = 0; tj < 2; ++tj)
          acc[ti][tj] = __builtin_amdgcn_wmma_f32_16x16x4_f32(
              /*neg_a=*/false, a[ti], /*neg_b=*/false, b[tj],
              /*c_mod=*/(short)0, acc[ti][tj],
              /*reuse_a=*/false, /*reuse_b=*/false);
    }

    // Swap buffers: my DMA must be done, and everyone must be past compute.
    if (s + 1 < NSLAB) {
      wait_async0();
      __syncthreads();
    }
  }

  // Epilogue: out = (xsq + ysq) * (1/d) - (2/d) * dot
  const float inv_d   = 1.0f / (float)K_DIM;
  const float neg2inv = -2.0f * inv_d;

  float ysq_v[2];
#pragma unroll
  for (int tj = 0; tj < 2; ++tj)
    ysq_v[tj] = ysq[colBase + wc * 32 + tj * 16 + lmod];

#pragma unroll
  for (int ti = 0; ti < 4; ++ti) {
#pragma unroll
    for (int r = 0; r < 8; ++r) {
      int row = rowBase + wr * 64 + ti * 16 + r + 8 * lhalf;  // C/D: M = r + 8*(lane/16)
      float xv = xsq[row];
#pragma unroll
      for (int tj = 0; tj < 2; ++tj) {
        int col = colBase + wc * 32 + tj * 16 + lmod;         // C/D: N = lane%16
        out[(size_t)row * Mcols + col] =
            fmaf(acc[ti][tj][r], neg2inv, (xv + ysq_v[tj]) * inv_d);
      }
    }
  }
}

// ---------------------------------------------------------------------------
extern "C" void kernel_launch(void* const* d_in, const int* in_sizes, int n_in,
                              void* d_out, int out_size, void* d_ws, size_t ws_size,
                              hipStream_t stream) {
  (void)n_in; (void)out_size; (void)ws_size;

  const float* X = (const float*)d_in[0];   // (N, 512) queries
  const float* Y = (const float*)d_in[1];   // (M, 512) prototypes
  float* out = (float*)d_out;

  const int N = in_sizes[0] / K_DIM;        // 8192
  const int M = in_sizes[1] / K_DIM;        // 4096

  float* xsq = (float*)d_ws;                // N floats
  float* ysq = xsq + N;                     // M floats

  // Norm kernel: one wave per row.
  int totalRows  = N + M;
  int normBlocks = (totalRows * 32 + 255) / 256;
  row_sqnorm_kernel<<<normBlocks, 256, 0, stream>>>(X, Y, xsq, ysq, N, M);

  // WMMA distance GEMM: 128x128 tiles.
  dim3 grid((unsigned)(M / 128), (unsigned)(N / 128));
  dist_wmma_kernel<<<grid, 256, 0, stream>>>(X, Y, xsq, ysq, out, M);
}